// DepthEncoder_8899172237827
// MI455X (gfx1250) — compile-verified
//
#include <hip/hip_runtime.h>
#include <hip/hip_bf16.h>
#include <math.h>

typedef __attribute__((ext_vector_type(2))) float v2f;
typedef __attribute__((ext_vector_type(4))) float v4f;
typedef __attribute__((ext_vector_type(8))) float v8f;

#define NB   16
#define HH   1024
#define WW   1024
#define PSZ  32

// One wave (32 threads) per block. Each wave owns 4 horizontally-adjacent
// 32x32 patches (a 32x128 strip) and accumulates the 4x4 augmented moment
// matrix S = sum_n [x y z 1]^T [x y z 1] for each patch via
// V_WMMA_F32_16X16X4_F32 (4 patches packed block-diagonally in the 16x16
// tile; B == A by symmetry of the Gram product).
// K-slot mapping (order-invariant sum): lo lanes sweep pixels of patch rows
// 0..15, hi lanes rows 16..31; one b128 load feeds two WMMA steps.
__global__ __launch_bounds__(32)
void depth_encoder_kernel(const float* __restrict__ depth, float* __restrict__ out)
{
    const int L   = threadIdx.x;   // 0..31
    const int r16 = L & 15;        // WMMA row (A) / col (B) index
    const int hi  = L >> 4;        // 0: K slots 0,1   1: K slots 2,3
    const int ip  = r16 >> 2;      // patch within group (0..3)
    const int c   = r16 & 3;       // component 0=x 1=y 2=z 3=1

    const int pg  = blockIdx.x;    // wave-group id, 4096 total
    const int b   = pg >> 8;       // batch (256 groups of 4 patches per batch)
    const int g   = pg & 255;
    const int p1  = g >> 3;        // patch row 0..31
    const int p2b = (g & 7) << 2;  // first patch col of this group

    const float fpx  = 886.8100134752651f;   // 512 / tan(30 deg)
    const float invf = 1.0f / fpx;
    const float cx = 512.0f, cy = 512.0f;

    const int colg = (p2b + ip) * PSZ;       // global col of this lane's patch
    const int rowg = p1 * PSZ;               // global row of patch origin

    const float* pbase = depth + ((size_t)b * HH + rowg) * WW + colg;
    const float* pLane = pbase + (size_t)(hi * 16) * WW;  // lo: rows 0-15, hi: 16-31

    const bool isX = (c == 0), isY = (c == 1), isZ = (c == 2), isW = (c == 3);

    // Per-patch shift = point at patch origin pixel (conditioning; cov is
    // shift-invariant). Homogeneous component keeps shift 0.
    float d00 = pbase[0];
    float a00 = isX ? ((float)colg - cx) * invf
              : isY ? ((float)rowg - cy) * invf
              : isZ ? 1.0f : 0.0f;
    float sshift = a00 * d00;
    float beta   = isW ? 1.0f : -sshift;

    // Branchless component coefficients: al = wx*ax + (isY ? ay : czw)
    const float wx     = isX ? 1.0f : 0.0f;
    const float czw    = isZ ? 1.0f : 0.0f;     // 0 for x,y,w lanes
    const float wxinvf = wx * invf;
    const float axBase = ((float)colg - cx) * invf;
    const float ayBase = ((float)(rowg + hi * 16) - cy) * invf;

    v8f acc0 = {0.f,0.f,0.f,0.f,0.f,0.f,0.f,0.f};
    v8f acc1 = {0.f,0.f,0.f,0.f,0.f,0.f,0.f,0.f};

    auto addrOf = [&](int jj) -> const v4f* {
        int rowl = jj >> 4;            // local row within this lane's half
        int col4 = (jj & 15) << 1;     // 0,4,...,28 -> 16B aligned
        return reinterpret_cast<const v4f*>(pLane + (size_t)rowl * WW + col4);
    };

    v4f cur = *addrOf(0);
    #pragma unroll 4
    for (int jj = 0; jj < 256; jj += 2) {
        int pf = jj + 2;
        pf = (pf > 254) ? 254 : pf;          // clamp: tail re-reads in-bounds
        v4f nxt = *addrOf(pf);               // prefetch next iteration's line

        float col4f = (float)((jj & 15) << 1);
        float rowlf = (float)(jj >> 4);
        float ax0 = fmaf(col4f, invf, axBase);
        float ay  = fmaf(rowlf, invf, ayBase);
        float base = isY ? ay : czw;

        float al0 = fmaf(wx, ax0, base);
        float al1 = al0 + wxinvf;
        float al2 = al1 + wxinvf;
        float al3 = al2 + wxinvf;

        v2f A0, A1;
        A0.x = fmaf(al0, cur.x, beta);
        A0.y = fmaf(al1, cur.y, beta);
        A1.x = fmaf(al2, cur.z, beta);
        A1.y = fmaf(al3, cur.w, beta);

        acc0 = __builtin_amdgcn_wmma_f32_16x16x4_f32(false, A0, false, A0,
                                                     (short)0, acc0, false, false);
        acc1 = __builtin_amdgcn_wmma_f32_16x16x4_f32(false, A1, false, A1,
                                                     (short)0, acc1, false, false);
        cur = nxt;
    }
    v8f acc = acc0 + acc1;

    // Spill 16x16 D tile to LDS. D layout: VGPR v -> M=v (lanes 0-15, N=lane)
    // or M=v+8 (lanes 16-31, N=lane-16).
    __shared__ float Dm[256];
    __shared__ float Sh[16];
    #pragma unroll
    for (int v = 0; v < 8; ++v)
        Dm[(v + 8 * hi) * 16 + r16] = acc[v];
    if (hi == 0) Sh[r16] = sshift;   // Sh[4*patch + comp]
    __syncthreads();

    if (L < 4) {
        const int o = L * 4;   // this lane finalizes patch L of the group
        #define SD(a, bb) Dm[(o + (a)) * 16 + (o + (bb))]
        float S00 = SD(0,0), S01 = SD(0,1), S02 = SD(0,2), S03 = SD(0,3);
        float S11 = SD(1,1), S12 = SD(1,2), S13 = SD(1,3);
        float S22 = SD(2,2), S23 = SD(2,3);
        #undef SD
        const float n = 1024.0f, invn = 1.0f / 1024.0f;
        const float invne = 1.0f / (n + 1e-6f);

        float mx = S03 * invn, my = S13 * invn, mz = S23 * invn; // shifted means
        float meanx = mx + Sh[o + 0];
        float meany = my + Sh[o + 1];
        float meanz = mz + Sh[o + 2];

        float c00 = (S00 - S03 * mx) * invne;
        float c01 = (S01 - S03 * my) * invne;
        float c02 = (S02 - S03 * mz) * invne;
        float c11 = (S11 - S13 * my) * invne;
        float c12 = (S12 - S13 * mz) * invne;
        float c22 = (S22 - S23 * mz) * invne;
        float dvar = (S22 - S23 * mz) * invn;   // jnp.var (ddof=0)

        // Closed-form symmetric 3x3 eigensolve (trigonometric method)
        float q  = (c00 + c11 + c22) * (1.0f / 3.0f);
        float pp = c01*c01 + c02*c02 + c12*c12;
        float b00 = c00 - q, b11 = c11 - q, b22 = c22 - q;
        float p2 = b00*b00 + b11*b11 + b22*b22 + 2.0f * pp;
        float e1, e2, e3, nx, ny, nz;
        if (p2 <= 1e-28f) {
            e1 = e2 = e3 = q; nx = 0.0f; ny = 0.0f; nz = 1.0f;
        } else {
            float p    = sqrtf(p2 * (1.0f / 6.0f));
            float invp = 1.0f / p;
            float B00 = b00*invp, B11 = b11*invp, B22 = b22*invp;
            float B01 = c01*invp, B02 = c02*invp, B12 = c12*invp;
            float detB = B00*(B11*B22 - B12*B12)
                       - B01*(B01*B22 - B12*B02)
                       + B02*(B01*B12 - B11*B02);
            float r = fminf(1.0f, fmaxf(-1.0f, 0.5f * detB));
            float phi = acosf(r) * (1.0f / 3.0f);
            e3 = q + 2.0f * p * cosf(phi);
            e1 = q + 2.0f * p * cosf(phi + 2.0943951023931953f);
            e2 = 3.0f * q - e1 - e3;
            // eigvec(e1) via Cayley-Hamilton: M = C^2 - (e2+e3)C + e2*e3*I
            float s = e2 + e3, t = e2 * e3;
            float A00 = c00*c00 + c01*c01 + c02*c02;
            float A01 = c00*c01 + c01*c11 + c02*c12;
            float A02 = c00*c02 + c01*c12 + c02*c22;
            float A11 = c01*c01 + c11*c11 + c12*c12;
            float A12 = c01*c02 + c11*c12 + c12*c22;
            float A22 = c02*c02 + c12*c12 + c22*c22;
            float M00 = A00 - s*c00 + t, M01 = A01 - s*c01, M02 = A02 - s*c02;
            float M11 = A11 - s*c11 + t, M12 = A12 - s*c12, M22 = A22 - s*c22 + t;
            float n0 = M00*M00 + M01*M01 + M02*M02;
            float n1 = M01*M01 + M11*M11 + M12*M12;
            float n2 = M02*M02 + M12*M12 + M22*M22;
            float vx, vy, vz;
            if (n0 >= n1 && n0 >= n2)      { vx = M00; vy = M01; vz = M02; }
            else if (n1 >= n2)             { vx = M01; vy = M11; vz = M12; }
            else                           { vx = M02; vy = M12; vz = M22; }
            float nn = sqrtf(vx*vx + vy*vy + vz*vz);
            if (nn < 1e-20f) { nx = 0.0f; ny = 0.0f; nz = 1.0f; }
            else {
                float inv = 1.0f / (nn * (1.0f + 1e-6f));
                nx = vx * inv; ny = vy * inv; nz = vz * inv;
                // deterministic sign (eigh sign is arbitrary anyway)
                float ax = fabsf(nx), ayv = fabsf(ny), az = fabsf(nz);
                float mmax = fmaxf(ax, fmaxf(ayv, az));
                float lead = (mmax == ax) ? nx : ((mmax == ayv) ? ny : nz);
                if (lead < 0.0f) { nx = -nx; ny = -ny; nz = -nz; }
            }
        }
        float planarity = e2 / (e3 + 1e-6f);

        size_t obase = ((size_t)b * 1024 + (size_t)(p1 * 32 + p2b + L)) * 8;
        out[obase + 0] = meanx;
        out[obase + 1] = meany;
        out[obase + 2] = meanz;
        out[obase + 3] = nx;
        out[obase + 4] = ny;
        out[obase + 5] = nz;
        out[obase + 6] = dvar;
        out[obase + 7] = planarity;
    }
}

extern "C" void kernel_launch(void* const* d_in, const int* in_sizes, int n_in,
                              void* d_out, int out_size, void* d_ws, size_t ws_size,
                              hipStream_t stream) {
    (void)in_sizes; (void)n_in; (void)out_size; (void)d_ws; (void)ws_size;
    const float* depth = (const float*)d_in[0];
    float* out = (float*)d_out;
    // 16 batches * 1024 patches / 4 patches-per-wave = 4096 waves
    depth_encoder_kernel<<<dim3(4096), dim3(32), 0, stream>>>(depth, out);
}